// GCN_35021163331781
// MI455X (gfx1250) — compile-verified
//
#include <hip/hip_runtime.h>

#define FEAT 64

typedef float v2f __attribute__((ext_vector_type(2)));
typedef float v8f __attribute__((ext_vector_type(8)));

// CDNA5 no-return FP32 global atomic (RMW resolved at L2 atomic units).
__device__ __forceinline__ void atomic_add_f32(float* p, float v) {
    asm volatile("global_atomic_add_f32 %0, %1, off" : : "v"(p), "v"(v) : "memory");
}

__global__ __launch_bounds__(256) void zero_f32x4(float4* __restrict__ p, int n4) {
    int i = blockIdx.x * blockDim.x + threadIdx.x;
    if (i < n4) p[i] = make_float4(0.f, 0.f, 0.f, 0.f);
}

// One wave32 per edge: each lane moves a float2 slice of the 64-float feature row.
// Gather is a fully-coalesced 256B segment; scatter-add is 64 f32 atomics per edge.
__global__ __launch_bounds__(256) void scatter_add64(
    const float* __restrict__ x, const int* __restrict__ src,
    const int* __restrict__ dst, float* __restrict__ agg, int nedges)
{
    int gid  = blockIdx.x * blockDim.x + threadIdx.x;
    int e    = gid >> 5;
    int lane = gid & 31;
    if (e >= nedges) return;
    int s = src[e];
    int d = dst[e];
    float2 v = *(const float2*)(x + (size_t)s * FEAT + lane * 2);
    float* ap = agg + (size_t)d * FEAT + lane * 2;
    atomic_add_f32(ap, v.x);
    atomic_add_f32(ap + 1, v.y);
}

// out[nrows x OUTF] = (relu?)(A[nrows x 64] @ W[64 x OUTF] + bias)
// One wave per 16x16 output tile, accumulated with V_WMMA_F32_16X16X4_F32.
//
// Register layouts per CDNA5 ISA 7.12.2 (32-bit, wave32):
//   A 16x4 : VGPR0 = {lanes0-15: K=k0, lanes16-31: K=k0+2}, VGPR1 = K+1
//            -> lane loads contiguous v2f at A[row][k0 + 2*(lane>>4)]
//   B 4x16 : VGPR0 = {lanes0-15: K=k0 row, lanes16-31: K=k0+2 row}, VGPR1 = K+1
//            -> lane loads W[k0+2*(lane>>4) + {0,1}][col]
//   D 16x16: VGPR v = {lanes0-15: M=v, lanes16-31: M=v+8}, N = lane&15
template<int OUTF, bool RELU>
__global__ __launch_bounds__(256) void gemm_bias_relu(
    const float* __restrict__ A, const float* __restrict__ W,
    const float* __restrict__ bias, float* __restrict__ out, int nrows)
{
    constexpr int NT = OUTF / 16;
    int wave = (blockIdx.x * blockDim.x + threadIdx.x) >> 5;
    int lane = threadIdx.x & 31;
    int tileM = wave / NT;
    int tileN = wave % NT;
    int row0 = tileM * 16;
    if (row0 >= nrows) return;
    int col0 = tileN * 16;
    int half = lane >> 4;   // 0 or 1: selects K pair
    int idx  = lane & 15;   // A row / B,D column within tile

    v8f c = {0.f, 0.f, 0.f, 0.f, 0.f, 0.f, 0.f, 0.f};
    const float* arow = A + (size_t)(row0 + idx) * FEAT + 2 * half;
    const float* wcol = W + col0 + idx;

#pragma unroll
    for (int k0 = 0; k0 < FEAT; k0 += 4) {
        v2f a = *(const v2f*)(arow + k0);
        v2f b;
        b.x = wcol[(k0 + 2 * half + 0) * OUTF];
        b.y = wcol[(k0 + 2 * half + 1) * OUTF];
        c = __builtin_amdgcn_wmma_f32_16x16x4_f32(
                /*neg_a=*/false, a, /*neg_b=*/false, b,
                /*c_mod=*/(short)0, c, /*reuse_a=*/false, /*reuse_b=*/false);
    }

    float bv = bias[col0 + idx];
#pragma unroll
    for (int v = 0; v < 8; ++v) {
        int m = v + 8 * half;
        float val = c[v] + bv;
        if (RELU) val = val > 0.f ? val : 0.f;
        out[(size_t)(row0 + m) * OUTF + col0 + idx] = val;
    }
}

extern "C" void kernel_launch(void* const* d_in, const int* in_sizes, int n_in,
                              void* d_out, int out_size, void* d_ws, size_t ws_size,
                              hipStream_t stream) {
    const float* features = (const float*)d_in[0];
    const int*   src      = (const int*)d_in[1];
    const int*   dst      = (const int*)d_in[2];
    const float* W1       = (const float*)d_in[3];
    const float* b1       = (const float*)d_in[4];
    const float* W2       = (const float*)d_in[5];
    const float* b2       = (const float*)d_in[6];
    const float* Wfc      = (const float*)d_in[7];
    const float* bfc      = (const float*)d_in[8];
    float* out = (float*)d_out;

    int nnodes = in_sizes[0] / FEAT;
    int nedges = in_sizes[1];

    float* buf0 = (float*)d_ws;                       // 25.6 MB agg buffer
    float* buf1 = buf0 + (size_t)nnodes * FEAT;       // 25.6 MB hidden buffer

    int n4 = (nnodes * FEAT) / 4;
    int zeroBlocks    = (n4 + 255) / 256;
    int scatterBlocks = (nedges + 7) / 8;             // 8 waves (edges) per block
    int tilesM = (nnodes + 15) / 16;
    int gemm64Blocks = (tilesM * 4 * 32 + 255) / 256; // 4 col tiles
    int gemm32Blocks = (tilesM * 2 * 32 + 255) / 256; // 2 col tiles
    dim3 blk(256);

    // hop 1: agg = scatter_sum(features); h1 = relu(agg @ W1 + b1)
    zero_f32x4<<<zeroBlocks, blk, 0, stream>>>((float4*)buf0, n4);
    scatter_add64<<<scatterBlocks, blk, 0, stream>>>(features, src, dst, buf0, nedges);
    gemm_bias_relu<64, true><<<gemm64Blocks, blk, 0, stream>>>(buf0, W1, b1, buf1, nnodes);

    // hop 2: agg = scatter_sum(h1); h2 = relu(agg @ W2 + b2)
    zero_f32x4<<<zeroBlocks, blk, 0, stream>>>((float4*)buf0, n4);
    scatter_add64<<<scatterBlocks, blk, 0, stream>>>(buf1, src, dst, buf0, nedges);
    gemm_bias_relu<64, true><<<gemm64Blocks, blk, 0, stream>>>(buf0, W2, b2, buf1, nnodes);

    // logits = h2 @ Wfc + bfc
    gemm_bias_relu<32, false><<<gemm32Blocks, blk, 0, stream>>>(buf1, Wfc, bfc, out, nnodes);
}